// Up_68899865362703
// MI455X (gfx1250) — compile-verified
//
#include <hip/hip_runtime.h>
#include <math.h>

// MI455X / gfx1250 implementation.
// Dominant cost: grid1 = 212MB of f32 stores (HBM-bound, ~9us floor at 23.3TB/s).
// The 9-deep Y-contraction per output element is offloaded to V_WMMA_F32_16X16X4_F32
// (FP32 WMMA, matches reference precision), K padded 9->12 (3 k-steps of 4).

typedef __attribute__((ext_vector_type(2))) float v2f;
typedef __attribute__((ext_vector_type(8))) float v8f;

#define NK1   11520           // N_RADII * idim(HID1)
#define NVOX  64
#define GPTS  72              // 6 * 12 sphere grid
#define NPAD  80              // GPTS padded to 5 * 16
#define KTILE 256             // k rows per workgroup

// -------------------- prep: tiny equivariant linears + Y + radii --------------------
__global__ __launch_bounds__(512) void prep_kernel(
    const float* __restrict__ x,
    const float* __restrict__ w1b0_0e, const float* __restrict__ w1b0_1o,
    const float* __restrict__ w1b0_2e,
    const float* __restrict__ w1b1_0e, const float* __restrict__ w1b1_1o,
    const float* __restrict__ w1b1_2e,
    float* __restrict__ xout,     // d_out[0 : 64*180]
    float* __restrict__ radii,    // d_out tail, 512 floats
    float* __restrict__ csrc,     // ws: [64][96] = y0e(32) | y1o(24) | y2e(40)
    float* __restrict__ ypad)     // ws: [12][80] zero-padded Y
{
  const int t = threadIdx.x;
  const float s32 = 0.17677669529663687f;   // 1/sqrt(32)
  const float s8  = 0.35355339059327373f;   // 1/sqrt(8)
  const float PI  = 3.14159265358979323846f;

  if (t < NVOX) {
    const int v = t;
    const float* xv = x + v * 120;
    float y0e[32];
    float y1o[8][3];
    float y2e[8][5];
    for (int o = 0; o < 32; ++o) y0e[o] = 0.f;
    for (int o = 0; o < 8; ++o) { for (int j = 0; j < 3; ++j) y1o[o][j] = 0.f;
                                  for (int j = 0; j < 5; ++j) y2e[o][j] = 0.f; }
    // 0e: x[0:32] -(w1b0_0e 32x64, /sqrt32)-> h -(w1b1_0e 64x32, /8)-> y0e
    for (int i = 0; i < 64; ++i) {
      float h = 0.f;
      for (int q = 0; q < 32; ++q) h += xv[q] * w1b0_0e[q * 64 + i];
      h *= s32;
      for (int o = 0; o < 32; ++o) y0e[o] += h * w1b1_0e[i * 32 + o];
    }
    for (int o = 0; o < 32; ++o) y0e[o] *= 0.125f;
    // 1o: x[32:80] (16x3) -(w1b0_1o 16x16, /4)-> h -(w1b1_1o 16x8, /4)-> y1o
    for (int i = 0; i < 16; ++i) {
      float h[3] = {0.f, 0.f, 0.f};
      for (int q = 0; q < 16; ++q)
        for (int j = 0; j < 3; ++j) h[j] += xv[32 + q * 3 + j] * w1b0_1o[q * 16 + i];
      for (int j = 0; j < 3; ++j) h[j] *= 0.25f;
      for (int o = 0; o < 8; ++o)
        for (int j = 0; j < 3; ++j) y1o[o][j] += h[j] * w1b1_1o[i * 8 + o];
    }
    for (int o = 0; o < 8; ++o) for (int j = 0; j < 3; ++j) y1o[o][j] *= 0.25f;
    // 2e: x[80:120] (8x5) -(w1b0_2e 8x16, /sqrt8)-> h -(w1b1_2e 16x8, /4)-> y2e
    for (int i = 0; i < 16; ++i) {
      float h[5] = {0.f, 0.f, 0.f, 0.f, 0.f};
      for (int q = 0; q < 8; ++q)
        for (int j = 0; j < 5; ++j) h[j] += xv[80 + q * 5 + j] * w1b0_2e[q * 16 + i];
      for (int j = 0; j < 5; ++j) h[j] *= s8;
      for (int o = 0; o < 8; ++o)
        for (int j = 0; j < 5; ++j) y2e[o][j] += h[j] * w1b1_2e[i * 8 + o];
    }
    for (int o = 0; o < 8; ++o) for (int j = 0; j < 5; ++j) y2e[o][j] *= 0.25f;

    // x_out layout (180): [y0e 32 | zeros 64 | y2e 40 (o*5+j) | y1o 24 (o*3+j) | zeros 20]
    float* xo = xout + v * 180;
    float* cs = csrc + v * 96;
    for (int o = 0; o < 32; ++o) { xo[o] = y0e[o]; cs[o] = y0e[o]; }
    for (int d = 32; d < 96; ++d) xo[d] = 0.f;
    for (int o = 0; o < 8; ++o)
      for (int j = 0; j < 5; ++j) { xo[96 + o * 5 + j] = y2e[o][j]; cs[56 + o * 5 + j] = y2e[o][j]; }
    for (int o = 0; o < 8; ++o)
      for (int j = 0; j < 3; ++j) { xo[136 + o * 3 + j] = y1o[o][j]; cs[32 + o * 3 + j] = y1o[o][j]; }
    for (int d = 160; d < 180; ++d) xo[d] = 0.f;
  }

  // Ypad [12][80]: rows 9-11 and cols 72-79 are zero
  for (int i = t; i < 12 * NPAD; i += 512) {
    const int row = i / NPAD, col = i % NPAD;
    float val = 0.f;
    if (row < 9 && col < GPTS) {
      const int bi = col / 12, ai = col % 12;
      const float B = ((float)bi + 0.5f) * (PI / 6.0f);
      const float A = (float)ai * (PI / 6.0f);
      const float sB = sinf(B), cB = cosf(B), sA = sinf(A), cA = cosf(A);
      const float sx_ = sB * cA, sy_ = sB * sA, sz_ = cB;
      const float s3 = 1.7320508075688772f, s15 = 3.872983346207417f, s5 = 2.23606797749979f;
      switch (row) {
        case 0: val = 1.f; break;
        case 1: val = s3 * sx_; break;
        case 2: val = s3 * sy_; break;
        case 3: val = s3 * sz_; break;
        case 4: val = s15 * sx_ * sy_; break;
        case 5: val = s15 * sy_ * sz_; break;
        case 6: val = 0.5f * s5 * (3.f * sz_ * sz_ - 1.f); break;
        case 7: val = s15 * sx_ * sz_; break;
        case 8: val = 0.5f * s15 * (sx_ * sx_ - sy_ * sy_); break;
      }
    }
    ypad[i] = val;
  }

  // closest_radii: fine 8x8x8 grid, SCALE=2; argmin over bins linspace(0,2,64)
  for (int i = t; i < 512; i += 512) {
    const int cc[3] = { i >> 6, (i >> 3) & 7, i & 7 };
    float r2 = 0.f;
    for (int d = 0; d < 3; ++d) {
      const float fc = (float)cc[d];
      const float ctr = floorf(fc * 0.5f) * 2.0f + 0.5f;
      const float dd = fc - ctr;
      r2 += dd * dd;
    }
    const float r = sqrtf(r2);
    int best = 0; float bd = fabsf(r);
    for (int j = 1; j < 64; ++j) {
      const float dj = fabsf(r - (float)j * (2.0f / 63.0f));
      if (dj < bd) { bd = dj; best = j; }   // strict < : first-min like jnp.argmin
    }
    radii[i] = (float)best;
  }
}

// -------------------- grid1: c = x_small * W_big, grid1 = c * Y via f32 WMMA ----------
__global__ __launch_bounds__(256) void grid1_kernel(
    const float* __restrict__ w0e,   // [32][NK1]
    const float* __restrict__ w1o,   // [8][NK1]
    const float* __restrict__ w2e,   // [8][NK1]
    const float* __restrict__ csrc,  // [64][96]
    const float* __restrict__ ypad,  // [12][80]
    float* __restrict__ out)         // grid1 base: [64][NK1][72]
{
  __shared__ float sY[12 * NPAD];       // 3840 B
  __shared__ float sC[KTILE * 13];      // 13312 B, stride 13 -> conflict-free
  __shared__ float sx[96];

  const int t  = threadIdx.x;
  const int v  = blockIdx.y;
  const int k0 = blockIdx.x * KTILE;

  for (int i = t; i < 12 * NPAD; i += 256) sY[i] = ypad[i];
  if (t < 96) sx[t] = csrc[v * 96 + t];
  __syncthreads();

  // Each thread builds c[k0+t, 0:12] (zero-padded K) from L2-resident weights.
  {
    const int k = k0 + t;
    const float s32 = 0.17677669529663687f;
    const float s8  = 0.35355339059327373f;
    float c[12];
    float acc = 0.f;
    #pragma unroll
    for (int i = 0; i < 32; ++i) acc += sx[i] * w0e[i * NK1 + k];
    c[0] = acc * s32;
    #pragma unroll
    for (int j = 0; j < 3; ++j) {
      acc = 0.f;
      #pragma unroll
      for (int i = 0; i < 8; ++i) acc += sx[32 + i * 3 + j] * w1o[i * NK1 + k];
      c[1 + j] = acc * s8;
    }
    #pragma unroll
    for (int j = 0; j < 5; ++j) {
      acc = 0.f;
      #pragma unroll
      for (int i = 0; i < 8; ++i) acc += sx[56 + i * 5 + j] * w2e[i * NK1 + k];
      c[4 + j] = acc * s8;
    }
    c[9] = c[10] = c[11] = 0.f;
    #pragma unroll
    for (int q = 0; q < 12; ++q) sC[t * 13 + q] = c[q];
  }
  __syncthreads();

  // WMMA: wave handles 2 x (16k x 80g) tiles; 3 K-steps x 5 N-tiles each.
  const int wave  = t >> 5;
  const int lane  = t & 31;
  const int lhalf = lane >> 4;        // 0: lanes 0-15, 1: lanes 16-31
  const int l16   = lane & 15;

  for (int sub = 0; sub < 2; ++sub) {
    const int krow0 = wave * 32 + sub * 16;
    // A[16x4] frag per K-step: lane<16 holds A[m][kb],A[m][kb+1]; lane>=16 kb+2,kb+3
    v2f a[3];
    #pragma unroll
    for (int kc = 0; kc < 3; ++kc) {
      const int kb = kc * 4 + lhalf * 2;
      const float* row = &sC[(krow0 + l16) * 13];
      a[kc].x = row[kb];
      a[kc].y = row[kb + 1];
    }
    #pragma unroll
    for (int nt = 0; nt < 5; ++nt) {
      const int n0 = nt * 16;
      v8f d = {};
      #pragma unroll
      for (int kc = 0; kc < 3; ++kc) {
        const int kb = kc * 4 + lhalf * 2;
        v2f b;
        b.x = sY[(kb + 0) * NPAD + n0 + l16];   // B[4x16]: row kb, col n0+l16
        b.y = sY[(kb + 1) * NPAD + n0 + l16];
        d = __builtin_amdgcn_wmma_f32_16x16x4_f32(
                false, a[kc], false, b, (short)0, d, false, false);
      }
      // D layout: vgpr r = row (r + 8*lhalf), col = n0 + l16
      const int gcol = n0 + l16;
      if (gcol < GPTS) {
        const size_t base =
            ((size_t)v * NK1 + (size_t)(k0 + krow0 + 8 * lhalf)) * GPTS + gcol;
        #pragma unroll
        for (int r = 0; r < 8; ++r) out[base + (size_t)r * GPTS] = d[r];
      }
    }
  }
}

extern "C" void kernel_launch(void* const* d_in, const int* in_sizes, int n_in,
                              void* d_out, int out_size, void* d_ws, size_t ws_size,
                              hipStream_t stream) {
  const float* x        = (const float*)d_in[0];
  const float* w1b0_0e  = (const float*)d_in[1];
  const float* w1b0_1o  = (const float*)d_in[2];
  const float* w1b0_2e  = (const float*)d_in[3];
  // d_in[4..6] = w2b0_* feed only the dead grid0 path -> unused
  const float* w1b1_0e  = (const float*)d_in[7];
  // d_in[8], d_in[9], d_in[12] multiply exact-zero segments -> unused
  const float* w1b1_1o  = (const float*)d_in[10];
  const float* w1b1_2e  = (const float*)d_in[11];
  const float* w2b1_0e  = (const float*)d_in[13];
  const float* w2b1_1o  = (const float*)d_in[14];
  const float* w2b1_2e  = (const float*)d_in[15];

  float* out   = (float*)d_out;
  float* xout  = out;                                     // 64*180
  float* grid1 = out + (size_t)NVOX * 180;                // 64*11520*72
  float* radii = grid1 + (size_t)NVOX * NK1 * GPTS;       // 512

  float* csrc = (float*)d_ws;                             // 64*96 floats
  float* ypad = csrc + NVOX * 96;                         // 12*80 floats

  prep_kernel<<<dim3(1), dim3(512), 0, stream>>>(
      x, w1b0_0e, w1b0_1o, w1b0_2e, w1b1_0e, w1b1_1o, w1b1_2e,
      xout, radii, csrc, ypad);

  dim3 grid(NK1 / KTILE, NVOX);   // 45 x 64
  grid1_kernel<<<grid, dim3(256), 0, stream>>>(
      w2b1_0e, w2b1_1o, w2b1_2e, csrc, ypad, grid1);
}